// Encoder_47665547051183
// MI455X (gfx1250) — compile-verified
//
#include <hip/hip_runtime.h>

typedef __attribute__((ext_vector_type(2))) float v2f;
typedef __attribute__((ext_vector_type(8))) float v8f;

#define NB 16
#define NN 128
#define NT 8
#define NH 128
#define NCELLS (NB * NN * NN) /* 262144 edge cells */

// ---------------------------------------------------------------------------
// edge_hidden[b,i,j,h] = A*wA + adj*wadj + oh*wpih + (bA+badj+bpih)
// as D(16x16) = Arows(16x4) x W(4x16) per 16-cell x 16-channel tile,
// using V_WMMA_F32_16X16X4_F32 (exact f32, wave32).
// One wave handles 16 cells x all 128 channels (8 WMMAs).
// ---------------------------------------------------------------------------
__global__ __launch_bounds__(256) void edge_wmma_kernel(
    const float* __restrict__ A, const float* __restrict__ adj,
    const int* __restrict__ pi_h,
    const float* __restrict__ wA, const float* __restrict__ bA,
    const float* __restrict__ wadj, const float* __restrict__ badj,
    const float* __restrict__ wpih, const float* __restrict__ bpih,
    const int* __restrict__ hint_p,
    float* __restrict__ edge_out)
{
    const int lane = threadIdx.x & 31;
    const int wave = blockIdx.x * (blockDim.x >> 5) + (threadIdx.x >> 5);
    const int cell_base = wave * 16;
    const int hint = hint_p[0];              // uniform scalar load

    const int m  = lane & 15;                // M row within 16x16 tile
    const bool hi = lane >= 16;              // lanes 16-31 carry K=2,3

    const int cell = cell_base + m;          // flat (b,i,j)
    const int j = cell & (NN - 1);
    const int i = (cell >> 7) & (NN - 1);
    const int b = cell >> 14;

    // all lanes load (coalesced, hi half hits the same lines), then select
    const float av   = A[cell];
    const float adjv = adj[cell];
    const int   pv   = pi_h[(b * NT + hint) * NN + i];
    const float ohv  = (pv == j) ? 1.0f : 0.0f;

    // A-matrix 16x4 f32 layout: lanes 0-15 -> K=0 (v0), K=1 (v1);
    //                           lanes 16-31 -> K=2 (v0), K=3 (v1)
    v2f amat;
    amat.x = hi ? ohv  : av;
    amat.y = hi ? 1.0f : adjv;

    float* __restrict__ out_base = edge_out + (size_t)cell_base * NH;
    const int row_off = hi ? 8 : 0;          // D layout: lanes 16-31 hold M=8..15

    #pragma unroll
    for (int t = 0; t < NH / 16; ++t) {
        const int h = t * 16 + m;            // B-matrix column = lane%16
        const float w_lo_x = wA[h];
        const float w_lo_y = wadj[h];
        const float w_hi_x = wpih[h];
        const float w_hi_y = bA[h] + badj[h] + bpih[h];  // K=3 row carries fused bias
        v2f bmat;
        bmat.x = hi ? w_hi_x : w_lo_x;       // K=0 / K=2
        bmat.y = hi ? w_hi_y : w_lo_y;       // K=1 / K=3
        v8f c = {};
        c = __builtin_amdgcn_wmma_f32_16x16x4_f32(
                /*neg_a=*/false, amat, /*neg_b=*/false, bmat,
                /*c_mod=*/(short)0, c, /*reuse_a=*/false, /*reuse_b=*/false);
        // D: vgpr r, lane l -> row (r + 8*(l/16)), col = l%16
        #pragma unroll
        for (int r = 0; r < 8; ++r) {
            __builtin_nontemporal_store(c[r],
                out_base + (size_t)(r + row_off) * NH + h);
        }
    }
}

// ---------------------------------------------------------------------------
// node_hidden[b,n,h] and graph_hidden[b,h] (tiny, VALU)
// ---------------------------------------------------------------------------
__global__ __launch_bounds__(256) void node_graph_kernel(
    const float* __restrict__ pos, const float* __restrict__ s,
    const float* __restrict__ d_h,
    const float* __restrict__ wpos, const float* __restrict__ bpos,
    const float* __restrict__ ws, const float* __restrict__ bs,
    const float* __restrict__ wdh, const float* __restrict__ bdh,
    const float* __restrict__ gkey,
    const float* __restrict__ wg, const float* __restrict__ bg,
    const int* __restrict__ hint_p,
    float* __restrict__ node_out, float* __restrict__ graph_out)
{
    const int idx = blockIdx.x * blockDim.x + threadIdx.x;  // [0, B*N*H)
    const int hint = hint_p[0];
    const int h  = idx & (NH - 1);
    const int bn = idx >> 7;               // b*N + n
    const int n  = bn & (NN - 1);
    const int b  = bn >> 7;
    const float dv = d_h[(b * NT + hint) * NN + n];
    const float v = fmaf(pos[bn], wpos[h], bpos[h])
                  + fmaf(s[bn],   ws[h],   bs[h])
                  + fmaf(dv,      wdh[h],  bdh[h]);
    node_out[idx] = v;
    if (idx < NB * NH) {                   // graph_hidden piggybacks on first 2048 threads
        const int gb = idx >> 7;
        graph_out[idx] = fmaf(gkey[gb], wg[h], bg[h]);
    }
}

// ---------------------------------------------------------------------------
// adj_out[b,i,j] = ((i==j) | (adj+adjT>0) | (oh|ohT)) ? 1 : 0
// ---------------------------------------------------------------------------
__global__ __launch_bounds__(256) void adjout_kernel(
    const float* __restrict__ adj, const int* __restrict__ pi_h,
    const int* __restrict__ hint_p, float* __restrict__ adj_out)
{
    const int idx = blockIdx.x * blockDim.x + threadIdx.x;  // [0, B*N*N)
    const int hint = hint_p[0];
    const int j = idx & (NN - 1);
    const int i = (idx >> 7) & (NN - 1);
    const int b = idx >> 14;
    const float a_ij = adj[idx];
    const float a_ji = adj[(b * NN + j) * NN + i];
    const int pi_i = pi_h[(b * NT + hint) * NN + i];
    const int pi_j = pi_h[(b * NT + hint) * NN + j];
    const bool on = (i == j) || ((a_ij + a_ji) > 0.0f)
                 || (pi_i == j) || (pi_j == i);
    adj_out[idx] = on ? 1.0f : 0.0f;
}

extern "C" void kernel_launch(void* const* d_in, const int* in_sizes, int n_in,
                              void* d_out, int out_size, void* d_ws, size_t ws_size,
                              hipStream_t stream) {
    const float* pos   = (const float*)d_in[0];
    const float* sArr  = (const float*)d_in[1];
    const float* A     = (const float*)d_in[2];
    const float* adj   = (const float*)d_in[3];
    const float* d_h   = (const float*)d_in[4];
    const int*   pi_h  = (const int*)d_in[5];
    const float* gkey  = (const float*)d_in[6];
    const float* w_pos = (const float*)d_in[7],  *b_pos = (const float*)d_in[8];
    const float* w_s   = (const float*)d_in[9],  *b_s   = (const float*)d_in[10];
    const float* w_A   = (const float*)d_in[11], *b_A   = (const float*)d_in[12];
    const float* w_adj = (const float*)d_in[13], *b_adj = (const float*)d_in[14];
    const float* w_dh  = (const float*)d_in[15], *b_dh  = (const float*)d_in[16];
    const float* w_pih = (const float*)d_in[17], *b_pih = (const float*)d_in[18];
    const float* w_g   = (const float*)d_in[19], *b_g   = (const float*)d_in[20];
    const int*   hint  = (const int*)d_in[21];

    float* out   = (float*)d_out;
    float* node  = out;                                    // [B,N,H]
    float* edge  = node + (size_t)NB * NN * NH;            // [B,N,N,H]
    float* graph = edge + (size_t)NB * NN * NN * NH;       // [B,H]
    float* adjo  = graph + (size_t)NB * NH;                // [B,N,N]

    // edge: 262144 cells / 16 per wave = 16384 waves; 8 waves (256 thr) / block
    edge_wmma_kernel<<<NCELLS / 16 / 8, 256, 0, stream>>>(
        A, adj, pi_h, w_A, b_A, w_adj, b_adj, w_pih, b_pih, hint, edge);

    node_graph_kernel<<<(NB * NN * NH) / 256, 256, 0, stream>>>(
        pos, sArr, d_h, w_pos, b_pos, w_s, b_s, w_dh, b_dh,
        gkey, w_g, b_g, hint, node, graph);

    adjout_kernel<<<NCELLS / 256, 256, 0, stream>>>(adj, pi_h, hint, adjo);
}